// Rule_71107478553152
// MI455X (gfx1250) — compile-verified
//
#include <hip/hip_runtime.h>
#include <stdint.h>
#include <math.h>

// Problem constants (match the reference)
constexpr int kH  = 1024;
constexpr int kW  = 1024;
constexpr int kHW = kH * kW;
constexpr int kC  = 8;          // spin channels; channel 8 is 'b'
constexpr int kB  = 4;          // batches

// Tile config
constexpr int TY = 32;          // tile rows
constexpr int TX = 64;          // tile cols
constexpr int NT = 256;         // threads per block (8 wave32)
constexpr int RSTRIDE = 72;     // LDS row stride (floats): keeps b128 dests 16B-aligned
constexpr int RROWS   = 36;     // 2 halo + 32 + 2 halo
constexpr int BUFELTS = RROWS * RSTRIDE;      // 2592 floats
constexpr uint32_t BUFBYTES = BUFELTS * 4;    // 10368 B

typedef __attribute__((ext_vector_type(4))) float v4f;

// ---- CDNA5 async global->LDS DMA (ASYNCcnt-tracked), GVS addressing mode ----
__device__ __forceinline__ void async_load_lds_b128(uint32_t lds_addr,
                                                    const void* sbase,
                                                    uint32_t voff_bytes) {
  asm volatile("global_load_async_to_lds_b128 %0, %1, %2"
               :: "v"(lds_addr), "v"(voff_bytes), "s"(sbase) : "memory");
}
__device__ __forceinline__ void async_load_lds_b32(uint32_t lds_addr,
                                                   const void* sbase,
                                                   uint32_t voff_bytes) {
  asm volatile("global_load_async_to_lds_b32 %0, %1, %2"
               :: "v"(lds_addr), "v"(voff_bytes), "s"(sbase) : "memory");
}

// Wait until at most the newest channel's loads remain outstanding.
// Per-wave issue counts per channel: waves 0-1: 4, waves 2-3: 3, waves 4-7: 2.
// (If the compiler ever issues halo DMAs with EXEC=0 in other waves these
//  constants are conservative, never under-waiting: ordering stays correct.)
__device__ __forceinline__ void wait_async_keep_newest(int wid_sgpr) {
  if (wid_sgpr < 2)      asm volatile("s_wait_asynccnt 4" ::: "memory");
  else if (wid_sgpr < 4) asm volatile("s_wait_asynccnt 3" ::: "memory");
  else                   asm volatile("s_wait_asynccnt 2" ::: "memory");
}
__device__ __forceinline__ void wait_async_zero() {
  asm volatile("s_wait_asynccnt 0" ::: "memory");
}

__device__ __forceinline__ v4f ld4(const float* p)  { return *(const v4f*)p; }
__device__ __forceinline__ void st4(float* p, v4f v){ *(v4f*)p = v; }
// Non-temporal (streaming) global accesses: read-once / write-once streams.
__device__ __forceinline__ v4f ld4_nt(const float* p) {
  return __builtin_nontemporal_load((const v4f*)p);
}
__device__ __forceinline__ void st4_nt(float* p, v4f v) {
  __builtin_nontemporal_store(v, (v4f*)p);
}

// Issue one channel's cross-region DMA into the LDS buffer at lbase.
__device__ __forceinline__ void issue_channel(const float* ch, uint32_t lbase,
                                              int t, int by0, int bx0) {
  // center tile: 2 x b128 per thread (full EXEC)
  #pragma unroll
  for (int i = 0; i < 2; ++i) {
    const int v   = t + NT * i;              // 0..511 vec4 cells
    const int ly  = v >> 4;                  // 0..31
    const int lxv = v & 15;                  // x in units of 4
    const uint32_t lds  = lbase + (uint32_t)(((ly + 2) * RSTRIDE + 4 + 4 * lxv) * 4);
    const uint32_t goff = (uint32_t)((((by0 + ly) * kW) + bx0 + 4 * lxv) * 4);
    async_load_lds_b128(lds, ch, goff);
  }
  // row halo: rows -2,-1,TY,TY+1 (b128, wraps in y) -- waves 0,1 only (full EXEC)
  if (t < 64) {
    const int r   = t >> 4;                  // 0..3
    const int lxv = t & 15;
    const int row = (r < 2) ? r : 32 + r;    // 0,1,34,35
    const int ly  = (r < 2) ? r - 2 : 30 + r;
    const int gy  = (by0 + ly) & (kH - 1);
    const uint32_t lds  = lbase + (uint32_t)((row * RSTRIDE + 4 + 4 * lxv) * 4);
    const uint32_t goff = (uint32_t)(((gy * kW) + bx0 + 4 * lxv) * 4);
    async_load_lds_b128(lds, ch, goff);
  }
  // col halo: cols -2,-1,TX,TX+1 (b32, wraps in x) -- waves 0..3 only (full EXEC)
  if (t < 128) {
    const int ci  = t >> 5;                  // 0..3
    const int ly  = t & 31;
    const int col = (ci < 2) ? 2 + ci : 66 + ci;   // 2,3,68,69
    const int gxo = (ci < 2) ? ci - 2 : 62 + ci;   // -2,-1,64,65
    const int gx  = (bx0 + gxo) & (kW - 1);
    const uint32_t lds  = lbase + (uint32_t)((((ly + 2) * RSTRIDE) + col) * 4);
    const uint32_t goff = (uint32_t)((((by0 + ly) * kW) + gx) * 4);
    async_load_lds_b32(lds, ch, goff);
  }
}

__global__ __launch_bounds__(NT) void ising_sweep_kernel(
    const float* __restrict__ x,        // (4, 9, 1024, 1024)
    const float* __restrict__ rnd,      // (4, 8, 1024, 1024)
    const float* __restrict__ dropout,  // (1024, 1024)
    float* __restrict__ out)            // (4, 9, 1024, 1024)
{
  __shared__ __align__(16) float buf[2][BUFELTS];  // ping-pong raw-channel stage
  __shared__ __align__(16) float sc[BUFELTS];      // channel sum

  const int t   = threadIdx.x;
  const int b   = blockIdx.z;
  const int by0 = blockIdx.y * TY;
  const int bx0 = blockIdx.x * TX;
  const int wid = __builtin_amdgcn_readfirstlane(t >> 5);  // scalar wave id

  const float* xb = x   + (size_t)b * 9  * kHW;
  const float* rb = rnd + (size_t)b * kC * kHW;
  float*       ob = out + (size_t)b * 9  * kHW;

  // Low 32 bits of a flat LDS pointer == DS byte address (aperture rules).
  const uint32_t buf_base = (uint32_t)(uintptr_t)(&buf[0][0]);

  v4f sreg[kC][2];   // this thread's 2 float4 center groups x 8 channels

  // Prime the pipeline with channel 0.
  issue_channel(xb, buf_base, t, by0, bx0);

  #pragma unroll
  for (int c = 0; c < kC; ++c) {
    // Overlap: issue channel c+1 into the other buffer, then drain channel c.
    if (c < kC - 1) {
      issue_channel(xb + (size_t)(c + 1) * kHW,
                    buf_base + (uint32_t)((c + 1) & 1) * BUFBYTES, t, by0, bx0);
      wait_async_keep_newest(wid);   // async loads complete in order
    } else {
      wait_async_zero();
    }
    __syncthreads();                 // all waves' channel-c DMA visible

    const float* bufp = &buf[c & 1][0];

    // ---- accumulate channel c into sc (vectorized) + capture center spins ----
    #pragma unroll
    for (int i = 0; i < 2; ++i) {    // center: 2 float4 per thread
      const int v   = t + NT * i;
      const int idx = ((v >> 4) + 2) * RSTRIDE + (v & 15) * 4 + 4;
      const v4f bv  = ld4(bufp + idx);
      sreg[c][i] = bv;               // free capture: same cell phase B needs
      if (c == 0) st4(&sc[idx], bv);
      else        st4(&sc[idx], ld4(&sc[idx]) + bv);
    }
    if (t < 64) {                    // row halo: 1 float4
      const int r   = t >> 4;
      const int row = (r < 2) ? r : 32 + r;
      const int idx = row * RSTRIDE + (t & 15) * 4 + 4;
      const v4f bv  = ld4(bufp + idx);
      if (c == 0) st4(&sc[idx], bv);
      else        st4(&sc[idx], ld4(&sc[idx]) + bv);
    }
    if (t < 128) {                   // col halo: 1 scalar
      const int ci  = t >> 5;
      const int col = (ci < 2) ? 2 + ci : 66 + ci;
      const int idx = ((t & 31) + 2) * RSTRIDE + col;
      const float bv = bufp[idx];
      if (c == 0) sc[idx] = bv;
      else        sc[idx] += bv;
    }
    __syncthreads();                 // buf[c&1] reads done before channel c+2 lands
  }

  // ---- phase B: stencil + acceptance + flip, fully float4 ----
  #pragma unroll
  for (int i = 0; i < 2; ++i) {
    const int v   = t + NT * i;
    const int ly  = v >> 4;
    const int lx4 = (v & 15) * 4;
    const int gy  = by0 + ly;
    const int gx  = bx0 + lx4;
    const float* scp = &sc[(ly + 2) * RSTRIDE + (lx4 + 4)];

    const v4f ctr = ld4(scp);
    const v4f up1 = ld4(scp - RSTRIDE),     dn1 = ld4(scp + RSTRIDE);
    const v4f up2 = ld4(scp - 2 * RSTRIDE), dn2 = ld4(scp + 2 * RSTRIDE);
    const v4f lft = ld4(scp - 4),           rgt = ld4(scp + 4);
    // horizontal window w[k] = sc[cy][cx-2+k], k=0..7
    const float w[8] = { lft[2], lft[3], ctr[0], ctr[1],
                         ctr[2], ctr[3], rgt[0], rgt[1] };

    // Exact add order of the reference roll chain per element:
    // Js = sc; +=y-1,y+1; +=x-1,x+1; +=y-2,y+2; +=x-2,x+2
    float js[4];
    #pragma unroll
    for (int j = 0; j < 4; ++j) {
      float J = ctr[j];
      J += up1[j];   J += dn1[j];
      J += w[j + 1]; J += w[j + 3];
      J += up2[j];   J += dn2[j];
      J += w[j];     J += w[j + 4];
      js[j] = J;
    }

    const size_t pofs = (size_t)gy * kW + gx;
    const v4f bv4 = ld4_nt(xb + (size_t)kC * kHW + pofs);  // read-once stream
    const v4f dr4 = ld4(dropout + pofs);                   // reused by 4 batches: keep RT

    #pragma unroll
    for (int c = 0; c < kC; ++c) {
      const v4f rr = ld4_nt(rb + (size_t)c * kHW + pofs);  // read-once stream
      v4f o;
      #pragma unroll
      for (int j = 0; j < 4; ++j) {
        const float sv = sreg[c][i][j];
        const float de = 2.0f * sv * js[j];
        const float pe = expf(-de * bv4[j]);
        const float p  = (de <= 0.0f) ? 1.0f : pe;
        const bool  fl = (rr[j] < p) && (dr4[j] > 0.5f);
        o[j] = fl ? -sv : sv;
      }
      st4_nt(ob + (size_t)c * kHW + pofs, o);              // write-once stream
    }
    st4_nt(ob + (size_t)kC * kHW + pofs, bv4);             // pass-through b channel
  }
}

extern "C" void kernel_launch(void* const* d_in, const int* in_sizes, int n_in,
                              void* d_out, int out_size, void* d_ws, size_t ws_size,
                              hipStream_t stream) {
  const float* x       = (const float*)d_in[0];
  const float* rnd     = (const float*)d_in[1];
  const float* dropout = (const float*)d_in[2];
  float*       out     = (float*)d_out;

  dim3 grid(kW / TX, kH / TY, kB);   // (16, 32, 4)
  ising_sweep_kernel<<<grid, NT, 0, stream>>>(x, rnd, dropout, out);
}